// ParagraphGATGated_23965917512224
// MI455X (gfx1250) — compile-verified
//
#include <hip/hip_runtime.h>
#include <math.h>

// ---------------------------------------------------------------------------
// ParagraphGATGated — dual-graph GATv2 (heads 8,8,4) + sigmoid gate.
// GEMMs use V_WMMA_F32_16X16X4_F32 (full f32 precision matrix op on CDNA5),
// software-pipelined so next-step loads issue before the current WMMA group.
// Edge aggregation uses deterministic CSR gather (no fp atomics in hot path).
// ---------------------------------------------------------------------------

#define HID 128

typedef float v2f __attribute__((ext_vector_type(2)));
typedef float v8f __attribute__((ext_vector_type(8)));

__device__ __forceinline__ float wred_sum(float v) {
    #pragma unroll
    for (int m = 16; m > 0; m >>= 1) v += __shfl_xor(v, m, 32);
    return v;
}
__device__ __forceinline__ float wred_max(float v) {
    #pragma unroll
    for (int m = 16; m > 0; m >>= 1) v = fmaxf(v, __shfl_xor(v, m, 32));
    return v;
}

// ---------------------------------------------------------------------------
// Y[N, HC] = X[N,128] @ W[HC,128]^T + b      (HC = heads*128, multiple of 64)
// Block = 256 threads = 8 waves. Wave computes a 16-row x 64-col tile.
// WMMA f32 16x16x4: A lane(L): M=L&15, VGPR{0,1}=K{0,1}+2*(L>=16)
//                   B lane(L): N=L&15, VGPR{0,1}=K{0,1}+2*(L>=16)
//                   C/D VGPR r: lanes0-15 -> (M=r, N=lane); lanes16-31 -> (M=r+8)
// Software pipeline: loads for k-step i+1 are issued before the WMMAs of i.
// ---------------------------------------------------------------------------
__global__ __launch_bounds__(256) void gemm_bias_kernel(
    const float* __restrict__ X, const float* __restrict__ W,
    const float* __restrict__ b, float* __restrict__ Y, int N, int HC)
{
    const int lane    = threadIdx.x & 31;
    const int wave    = threadIdx.x >> 5;
    const int rowBase = blockIdx.x * 128 + wave * 16;
    if (rowBase >= N) return;                    // wave-uniform
    const int colBase = blockIdx.y * 64;
    const int half    = lane >> 4;               // 0 for lanes 0-15, 1 for 16-31
    const int l15     = lane & 15;
    const int aRow    = min(rowBase + l15, N - 1);

    v8f acc0, acc1, acc2, acc3;
    {
        const float b0 = b[colBase      + l15];
        const float b1 = b[colBase + 16 + l15];
        const float b2 = b[colBase + 32 + l15];
        const float b3 = b[colBase + 48 + l15];
        #pragma unroll
        for (int r = 0; r < 8; ++r) { acc0[r] = b0; acc1[r] = b1; acc2[r] = b2; acc3[r] = b3; }
    }

    const float* ap = X + (size_t)aRow * HID + 2 * half;
    const float* w0 = W + (size_t)(colBase      + l15) * HID + 2 * half;
    const float* w1 = W + (size_t)(colBase + 16 + l15) * HID + 2 * half;
    const float* w2 = W + (size_t)(colBase + 32 + l15) * HID + 2 * half;
    const float* w3 = W + (size_t)(colBase + 48 + l15) * HID + 2 * half;

    // Prologue: load k-step 0.
    v2f a  = *(const v2f*)(ap);
    v2f q0 = *(const v2f*)(w0);
    v2f q1 = *(const v2f*)(w1);
    v2f q2 = *(const v2f*)(w2);
    v2f q3 = *(const v2f*)(w3);

    #pragma unroll
    for (int k = 0; k < HID; k += 4) {
        v2f an, q0n, q1n, q2n, q3n;
        if (k + 4 < HID) {                       // compile-time resolvable
            const int kn = k + 4;
            an  = *(const v2f*)(ap + kn);
            q0n = *(const v2f*)(w0 + kn);
            q1n = *(const v2f*)(w1 + kn);
            q2n = *(const v2f*)(w2 + kn);
            q3n = *(const v2f*)(w3 + kn);
        }
        acc0 = __builtin_amdgcn_wmma_f32_16x16x4_f32(false, a, false, q0, (short)0, acc0, false, false);
        acc1 = __builtin_amdgcn_wmma_f32_16x16x4_f32(false, a, false, q1, (short)0, acc1, false, false);
        acc2 = __builtin_amdgcn_wmma_f32_16x16x4_f32(false, a, false, q2, (short)0, acc2, false, false);
        acc3 = __builtin_amdgcn_wmma_f32_16x16x4_f32(false, a, false, q3, (short)0, acc3, false, false);
        if (k + 4 < HID) {
            a = an; q0 = q0n; q1 = q1n; q2 = q2n; q3 = q3n;
        }
    }

    #pragma unroll
    for (int r = 0; r < 8; ++r) {
        const int row = rowBase + r + 8 * half;
        if (row < N) {
            float* yp = Y + (size_t)row * HC + colBase + l15;
            yp[0]  = acc0[r];
            yp[16] = acc1[r];
            yp[32] = acc2[r];
            yp[48] = acc3[r];
        }
    }
}

// ---------------------------------------------------------------------------
// CSR build over dst (edges e<E: src=ei[e], dst=ei[E+e]; e>=E: self loop e-E)
// ---------------------------------------------------------------------------
__global__ void count_deg_kernel(const int* __restrict__ ei, int E, int Etot,
                                 int* __restrict__ deg)
{
    const int e = blockIdx.x * 256 + threadIdx.x;
    if (e >= Etot) return;
    const int d = (e < E) ? ei[E + e] : (e - E);
    atomicAdd(&deg[d], 1);
}

__global__ __launch_bounds__(1024) void scan_kernel(const int* __restrict__ deg,
                                                    int N, int* __restrict__ rowptr)
{
    __shared__ int s[1024];
    __shared__ int s_run;
    const int tid = threadIdx.x;
    if (tid == 0) s_run = 0;
    __syncthreads();
    for (int base = 0; base < N; base += 1024) {
        const int v = (base + tid < N) ? deg[base + tid] : 0;
        s[tid] = v;
        __syncthreads();
        for (int off = 1; off < 1024; off <<= 1) {
            const int t = (tid >= off) ? s[tid - off] : 0;
            __syncthreads();
            s[tid] += t;
            __syncthreads();
        }
        const int run = s_run;
        if (base + tid < N) rowptr[base + tid] = run + s[tid] - v;   // exclusive
        __syncthreads();
        if (tid == 0) s_run = run + s[1023];
        __syncthreads();
    }
    if (tid == 0) rowptr[N] = s_run;
}

__global__ void fill_csr_kernel(const int* __restrict__ ei, int E, int Etot,
                                const int* __restrict__ rowptr,
                                int* __restrict__ cursor, int* __restrict__ csr_eid)
{
    const int e = blockIdx.x * 256 + threadIdx.x;
    if (e >= Etot) return;
    const int d   = (e < E) ? ei[E + e] : (e - E);
    const int pos = rowptr[d] + atomicAdd(&cursor[d], 1);
    csr_eid[pos] = e;
}

// Deterministic ordering: sort each segment by edge id (avg degree ~9).
__global__ void sort_csr_kernel(const int* __restrict__ rowptr, int N,
                                int* __restrict__ csr_eid)
{
    const int n = blockIdx.x * 256 + threadIdx.x;
    if (n >= N) return;
    const int b = rowptr[n], e = rowptr[n + 1];
    for (int i = b; i < e - 1; ++i) {
        int mi = i, mv = csr_eid[i];
        for (int j = i + 1; j < e; ++j) {
            const int v = csr_eid[j];
            if (v < mv) { mv = v; mi = j; }
        }
        if (mi != i) { csr_eid[mi] = csr_eid[i]; csr_eid[i] = mv; }
    }
}

// ---------------------------------------------------------------------------
// logits[e,h] = att[h,:] . leaky_relu(xl[src,h,:] + xr[dst,h,:], 0.2)
// One wave per edge; 32 lanes x float4 = 128 channels.
// ---------------------------------------------------------------------------
__global__ __launch_bounds__(256) void edge_logits_kernel(
    const int* __restrict__ ei, int E, int Etot,
    const float* __restrict__ xl, const float* __restrict__ xr,
    const float* __restrict__ att, int H, float* __restrict__ logits)
{
    const int lane = threadIdx.x & 31;
    const int e    = blockIdx.x * 8 + (threadIdx.x >> 5);
    if (e >= Etot) return;                       // wave-uniform
    int s, d;
    if (e < E) { s = ei[e]; d = ei[E + e]; } else { s = d = e - E; }
    const int c0 = lane * 4;
    for (int h = 0; h < H; ++h) {
        const float4 av = *(const float4*)(xl + ((size_t)s * H + h) * HID + c0);
        const float4 bv = *(const float4*)(xr + ((size_t)d * H + h) * HID + c0);
        const float4 at = *(const float4*)(att + (size_t)h * HID + c0);
        float vx = av.x + bv.x, vy = av.y + bv.y, vz = av.z + bv.z, vw = av.w + bv.w;
        vx = vx > 0.f ? vx : 0.2f * vx;
        vy = vy > 0.f ? vy : 0.2f * vy;
        vz = vz > 0.f ? vz : 0.2f * vz;
        vw = vw > 0.f ? vw : 0.2f * vw;
        float p = vx * at.x + vy * at.y + vz * at.z + vw * at.w;
        p = wred_sum(p);
        if (lane == 0) logits[(size_t)e * H + h] = p;
    }
}

// ---------------------------------------------------------------------------
// Per node: segment softmax over incoming edges per head, weighted gather of
// xl[src], mean over heads, +bias, +residual, optional ReLU. One wave/node.
// ---------------------------------------------------------------------------
__global__ __launch_bounds__(256) void node_aggregate_kernel(
    const int* __restrict__ rowptr, const int* __restrict__ csr_eid,
    const int* __restrict__ ei, int E,
    const float* __restrict__ logits, const float* __restrict__ xl,
    const float* __restrict__ x_old, const float* __restrict__ bias,
    int H, int N, int do_relu, float* __restrict__ x_new)
{
    const int lane = threadIdx.x & 31;
    const int n    = blockIdx.x * 8 + (threadIdx.x >> 5);
    if (n >= N) return;                          // wave-uniform
    const int beg = rowptr[n], end = rowptr[n + 1];
    const int c0  = lane * 4;

    float m0 = 0.f, m1 = 0.f, m2 = 0.f, m3 = 0.f;  // mean-over-heads accum

    for (int h = 0; h < H; ++h) {
        float lm = -INFINITY;
        for (int i = beg + lane; i < end; i += 32)
            lm = fmaxf(lm, logits[(size_t)csr_eid[i] * H + h]);
        const float m = wred_max(lm);            // finite: self-loop guarantees >=1 edge

        float lz = 0.f;
        for (int i = beg + lane; i < end; i += 32)
            lz += __expf(logits[(size_t)csr_eid[i] * H + h] - m);
        const float inv = 1.f / (wred_sum(lz) + 1e-16f);

        float a0 = 0.f, a1 = 0.f, a2 = 0.f, a3 = 0.f;
        for (int i = beg; i < end; ++i) {        // uniform serial loop (sorted order)
            const int e = csr_eid[i];
            const int s = (e < E) ? ei[e] : (e - E);
            const float alpha = __expf(logits[(size_t)e * H + h] - m) * inv;
            const float4 v = *(const float4*)(xl + ((size_t)s * H + h) * HID + c0);
            a0 = fmaf(alpha, v.x, a0);
            a1 = fmaf(alpha, v.y, a1);
            a2 = fmaf(alpha, v.z, a2);
            a3 = fmaf(alpha, v.w, a3);
        }
        m0 += a0; m1 += a1; m2 += a2; m3 += a3;
    }

    const float invH = 1.f / (float)H;
    const float4 bv = *(const float4*)(bias + c0);
    const float4 xo = *(const float4*)(x_old + (size_t)n * HID + c0);
    float4 r;
    r.x = m0 * invH + bv.x + xo.x;
    r.y = m1 * invH + bv.y + xo.y;
    r.z = m2 * invH + bv.z + xo.z;
    r.w = m3 * invH + bv.w + xo.w;
    if (do_relu) {
        r.x = fmaxf(r.x, 0.f); r.y = fmaxf(r.y, 0.f);
        r.z = fmaxf(r.z, 0.f); r.w = fmaxf(r.w, 0.f);
    }
    *(float4*)(x_new + (size_t)n * HID + c0) = r;
}

// ---------------------------------------------------------------------------
// out[p,:] = g*f1[p,:] + (1-g)*f2[p,:],  g = sigmoid([f1;f2].gate_W + gate_b)
// ---------------------------------------------------------------------------
__global__ __launch_bounds__(256) void gate_kernel(
    const float* __restrict__ f1, const float* __restrict__ f2,
    const float* __restrict__ gW, const float* __restrict__ gb,
    int NP, float* __restrict__ out)
{
    const int lane = threadIdx.x & 31;
    const int p    = blockIdx.x * 8 + (threadIdx.x >> 5);
    if (p >= NP) return;                         // wave-uniform
    const int c0 = lane * 4;
    const float4 a  = *(const float4*)(f1 + (size_t)p * HID + c0);
    const float4 c  = *(const float4*)(f2 + (size_t)p * HID + c0);
    const float4 w1 = *(const float4*)(gW + c0);
    const float4 w2 = *(const float4*)(gW + HID + c0);
    float s = a.x * w1.x + a.y * w1.y + a.z * w1.z + a.w * w1.w
            + c.x * w2.x + c.y * w2.y + c.z * w2.z + c.w * w2.w;
    s = wred_sum(s) + gb[0];
    const float g = 1.f / (1.f + __expf(-s));
    float4 o;
    o.x = g * a.x + (1.f - g) * c.x;
    o.y = g * a.y + (1.f - g) * c.y;
    o.z = g * a.z + (1.f - g) * c.z;
    o.w = g * a.w + (1.f - g) * c.w;
    *(float4*)(out + (size_t)p * HID + c0) = o;
}

// ---------------------------------------------------------------------------
// Host driver. Input flattening assumed in setup_inputs() insertion order:
//  0:x1  1:edge_index1  2:x2  3:edge_index2  4:num_paragraphs
//  5..22 : g1 layers 0..2, each {Wl,bl,Wr,br,att,bias}
//  23..40: g2 layers 0..2, same order
//  41:gate_W  42:gate_b
// ---------------------------------------------------------------------------
extern "C" void kernel_launch(void* const* d_in, const int* in_sizes, int n_in,
                              void* d_out, int out_size, void* d_ws, size_t ws_size,
                              hipStream_t stream)
{
    (void)n_in; (void)ws_size;
    const int N    = in_sizes[0] / HID;
    const int E    = in_sizes[1] / 2;
    const int Etot = E + N;
    const int NP   = out_size / HID;

    const float* x1  = (const float*)d_in[0];
    const int*   ei1 = (const int*)d_in[1];
    const float* x2  = (const float*)d_in[2];
    const int*   ei2 = (const int*)d_in[3];

    static const int HEADS[3] = {8, 8, 4};

    // Workspace carve (256B aligned): 3 x-buffers, xl, xr (sized for 8 heads),
    // logits, rowptr, deg/cursor, csr_eid.  ~203 MB total.
    char*  ws  = (char*)d_ws;
    size_t off = 0;
    #define WS_ALLOC(T, count) (T*)(ws + off); off += (((size_t)(count) * sizeof(T)) + 255) & ~(size_t)255
    float* xA      = WS_ALLOC(float, (size_t)N * HID);
    float* xB      = WS_ALLOC(float, (size_t)N * HID);
    float* xC      = WS_ALLOC(float, (size_t)N * HID);
    float* xl      = WS_ALLOC(float, (size_t)N * 8 * HID);
    float* xr      = WS_ALLOC(float, (size_t)N * 8 * HID);
    float* logits  = WS_ALLOC(float, (size_t)Etot * 8);
    int*   rowptr  = WS_ALLOC(int,   (size_t)N + 1);
    int*   deg     = WS_ALLOC(int,   (size_t)N);
    int*   csr_eid = WS_ALLOC(int,   (size_t)Etot);
    #undef WS_ALLOC

    auto run_graph = [&](const float* x_in, const int* ei, int pbase,
                         float* out0, float* out1, float* out2) -> float* {
        // --- CSR build (edge list is layer-invariant) ---
        hipMemsetAsync(deg, 0, (size_t)N * sizeof(int), stream);
        count_deg_kernel<<<(Etot + 255) / 256, 256, 0, stream>>>(ei, E, Etot, deg);
        scan_kernel<<<1, 1024, 0, stream>>>(deg, N, rowptr);
        hipMemsetAsync(deg, 0, (size_t)N * sizeof(int), stream);
        fill_csr_kernel<<<(Etot + 255) / 256, 256, 0, stream>>>(ei, E, Etot, rowptr, deg, csr_eid);
        sort_csr_kernel<<<(N + 255) / 256, 256, 0, stream>>>(rowptr, N, csr_eid);

        const float* xc = x_in;
        float* outs[3] = {out0, out1, out2};
        for (int l = 0; l < 3; ++l) {
            const int H  = HEADS[l];
            const int HC = H * HID;
            const float* Wl   = (const float*)d_in[pbase + l * 6 + 0];
            const float* bl   = (const float*)d_in[pbase + l * 6 + 1];
            const float* Wr   = (const float*)d_in[pbase + l * 6 + 2];
            const float* br   = (const float*)d_in[pbase + l * 6 + 3];
            const float* att  = (const float*)d_in[pbase + l * 6 + 4];
            const float* bias = (const float*)d_in[pbase + l * 6 + 5];

            dim3 gg((N + 127) / 128, HC / 64);
            gemm_bias_kernel<<<gg, 256, 0, stream>>>(xc, Wl, bl, xl, N, HC);
            gemm_bias_kernel<<<gg, 256, 0, stream>>>(xc, Wr, br, xr, N, HC);
            edge_logits_kernel<<<(Etot + 7) / 8, 256, 0, stream>>>(ei, E, Etot, xl, xr, att, H, logits);
            node_aggregate_kernel<<<(N + 7) / 8, 256, 0, stream>>>(
                rowptr, csr_eid, ei, E, logits, xl, xc, bias, H, N, (l < 2) ? 1 : 0, outs[l]);
            xc = outs[l];
        }
        return (float*)xc;
    };

    // Graph 1: x1 -> xA -> xB -> xA   (final = xA)
    float* f1 = run_graph(x1, ei1, 5, xA, xB, xA);
    // Graph 2: x2 -> xB -> xC -> xB   (final = xB; xA preserved)
    float* f2 = run_graph(x2, ei2, 5 + 18, xB, xC, xB);

    const float* gW = (const float*)d_in[5 + 36];
    const float* gb = (const float*)d_in[5 + 36 + 1];
    gate_kernel<<<(NP + 7) / 8, 256, 0, stream>>>(f1, f2, gW, gb, NP, (float*)d_out);
}